// DarkChannelLoss_16363825398434
// MI455X (gfx1250) — compile-verified
//
#include <hip/hip_runtime.h>

// ---------------------------------------------------------------------------
// Dark-channel MSE loss for MI455X (gfx1250).
// Bandwidth-bound (~100 MB in, scalar out -> ~4.3us floor @ 23.3 TB/s; whole
// working set fits in the 192 MB L2, so halo re-reads are L2 hits).
// Data movement: CDNA5 Tensor Data Mover (tensor_load_to_lds / TENSORcnt)
// DMAs strided f32 tiles into LDS -- 12 equal sub-loads per block, no
// duplicates. Erosion is separable 15-min in LDS; reduction is fixed-order
// (deterministic, no atomics). No WMMA: min-erosion has no matrix structure.
// ---------------------------------------------------------------------------

#define NB    16
#define HH    512
#define WW    512
#define TILE  32
#define PADR  7
#define HALO  (TILE + 2 * PADR)   // 46
#define HHALF 23                  // HALO/2 rows per TDM sub-load
#define NTHREADS 256
#define NBLOCKS  (16 * 16 * 16)   // 4096 partial sums

typedef unsigned int u32x4 __attribute__((ext_vector_type(4)));
typedef int          i32x4 __attribute__((ext_vector_type(4)));
typedef int          i32x8 __attribute__((ext_vector_type(8)));

// Issue one 2D TDM load: tile_w x tile_h f32 elements, row stride in elements.
// D# packing per CDNA5 ISA ch.8 (group0: count/type/lds/global addr,
// group1: data_size=4B, tensor dims == tile dims so OOB never clips).
__device__ __forceinline__ void tdm_load_tile_f32(const float* gsrc,
                                                  unsigned lds_byte,
                                                  int tile_w, int tile_h,
                                                  long long row_stride_elems) {
  unsigned long long ga = (unsigned long long)gsrc;
  u32x4 g0 = {
    1u,                                                  // count=1, user desc
    lds_byte,                                            // LDS byte address
    (unsigned)(ga & 0xFFFFFFFFull),                      // global_addr[31:0]
    (unsigned)((ga >> 32) & 0x01FFFFFFull) | (2u << 30)  // addr[56:32] | type=2
  };
  unsigned td0 = (unsigned)tile_w;
  unsigned td1 = (unsigned)tile_h;
  unsigned long long s0 = (unsigned long long)row_stride_elems;
  i32x8 g1 = {
    (int)(2u << 16),                                         // data_size = 4B
    (int)((td0 & 0xFFFFu) << 16),                            // tensor_dim0 lo16
    (int)(((td0 >> 16) & 0xFFFFu) | ((td1 & 0xFFFFu) << 16)),// td0 hi | td1 lo
    (int)(((td1 >> 16) & 0xFFFFu) | (td0 << 16)),            // td1 hi | tile_dim0
    (int)td1,                                                // tile_dim1 | tile_dim2=0
    (int)(unsigned)(s0 & 0xFFFFFFFFull),                     // dim0_stride lo32
    (int)(unsigned)((s0 >> 32) & 0xFFFFull),                 // dim0_stride hi16
    0
  };
  i32x4 g2 = {0, 0, 0, 0};
  i32x4 g3 = {0, 0, 0, 0};
#if defined(__clang_major__) && __clang_major__ >= 23
  i32x8 g4 = {0, 0, 0, 0, 0, 0, 0, 0};
  __builtin_amdgcn_tensor_load_to_lds(g0, g1, g2, g3, g4, 0);
#else
  __builtin_amdgcn_tensor_load_to_lds(g0, g1, g2, g3, 0);
#endif
}

__device__ __forceinline__ int reflect512(int g) {
  g = g < 0 ? -g : g;
  return g > (WW - 1) ? 2 * (WW - 1) - g : g;
}

__global__ __launch_bounds__(NTHREADS) void darkchannel_mse_tile(
    const float* __restrict__ real, const float* __restrict__ fake,
    float* __restrict__ partial) {
  __shared__ float sT[2][3][HALO][HALO];  // staged 3-channel halo tiles (both imgs)
  __shared__ float sR[HALO][TILE];        // row-min intermediate
  __shared__ float sRed[8];

  const int tid  = threadIdx.x;
  const int wid  = tid >> 5;
  const int lane = tid & 31;
  const int bz   = blockIdx.z;
  const int row0 = blockIdx.y * TILE;
  const int col0 = blockIdx.x * TILE;

  // Clamp halo fetch window so it is always fully in-bounds; reflected
  // indices provably land inside [start, start+HALO).
  int rstart = row0 - PADR;
  rstart = rstart < 0 ? 0 : rstart;
  rstart = rstart > HH - HALO ? HH - HALO : rstart;
  int cstart = col0 - PADR;
  cstart = cstart < 0 ? 0 : cstart;
  cstart = cstart > WW - HALO ? WW - HALO : cstart;

  // --- Stage 0: TDM DMA. 12 sub-loads (2 imgs x 3 ch x 2 row-halves), no
  // duplicates. Expressed as a divergent loop so waves without a second load
  // exit via s_cbranch_execz BEFORE reaching the (EXEC-ignoring) TDM issue.
  for (int L = wid; L < 12; L += 8) {
    int plane = L >> 1;               // 0..5
    int h     = L & 1;                // row half
    int img   = plane >= 3 ? 1 : 0;
    int ch    = plane - img * 3;
    const float* base = img == 0 ? real : fake;
    const float* src  = base + (long long)(bz * 3 + ch) * (HH * WW)
                             + (long long)(rstart + h * HHALF) * WW + cstart;
    unsigned lds = (unsigned)(unsigned long long)(const void*)&sT[img][ch][h * HHALF][0];
    tdm_load_tile_f32(src, lds, HALO, HHALF, WW);
  }
  __builtin_amdgcn_s_wait_tensorcnt(0);
  __syncthreads();

  // --- Stage 1: channel min + (x+1)/2 rescale, in place into channel 0.
  for (int y = wid; y < HALO; y += 8) {
    for (int x = lane; x < HALO; x += 32) {
#pragma unroll
      for (int i = 0; i < 2; ++i) {
        float m = fminf(sT[i][0][y][x], fminf(sT[i][1][y][x], sT[i][2][y][x]));
        sT[i][0][y][x] = (m + 1.0f) * 0.5f;
      }
    }
  }
  __syncthreads();

  // Per-thread reflected column offsets are loop-invariant: hoist to registers.
  int xoff[15];
#pragma unroll
  for (int d = 0; d < 15; ++d) xoff[d] = reflect512(col0 + lane + d - PADR) - cstart;

  float rmap[4];
  float acc = 0.0f;

#pragma unroll
  for (int img = 0; img < 2; ++img) {
    // --- Stage 2: horizontal 15-wide min with reflect indexing -> sR
    for (int y = wid; y < HALO; y += 8) {
      float m = sT[img][0][y][xoff[0]];
#pragma unroll
      for (int d = 1; d < 15; ++d) m = fminf(m, sT[img][0][y][xoff[d]]);
      sR[y][lane] = m;
    }
    __syncthreads();

    // --- Stage 3: vertical 15-tall min + clip; keep real's map in registers.
#pragma unroll
    for (int k = 0; k < 4; ++k) {
      int oy = wid + k * 8;
      float m = 1e30f;
#pragma unroll
      for (int d = -PADR; d <= PADR; ++d) {
        int gr = reflect512(row0 + oy + d) - rstart;
        m = fminf(m, sR[gr][lane]);
      }
      m = fminf(fmaxf(m, 0.0f), 0.1f);
      if (img == 0) {
        rmap[k] = m;
      } else {
        float df = rmap[k] - m;
        acc += df * df;
      }
    }
    __syncthreads();  // protect sR before reuse
  }

  // --- Deterministic block reduction (wave32 shfl -> LDS -> thread 0).
#pragma unroll
  for (int off = 16; off > 0; off >>= 1) acc += __shfl_down(acc, off, 32);
  if (lane == 0) sRed[wid] = acc;
  __syncthreads();
  if (tid == 0) {
    float s = 0.0f;
#pragma unroll
    for (int w = 0; w < 8; ++w) s += sRed[w];
    partial[(bz * 16 + (int)blockIdx.y) * 16 + (int)blockIdx.x] = s;
  }
}

__global__ __launch_bounds__(NTHREADS) void darkchannel_mse_finish(
    const float* __restrict__ partial, float* __restrict__ out) {
  __shared__ float sRed[8];
  int tid = threadIdx.x;
  float s = 0.0f;
#pragma unroll
  for (int i = 0; i < NBLOCKS / NTHREADS; ++i) s += partial[tid + i * NTHREADS];
#pragma unroll
  for (int off = 16; off > 0; off >>= 1) s += __shfl_down(s, off, 32);
  if ((tid & 31) == 0) sRed[tid >> 5] = s;
  __syncthreads();
  if (tid == 0) {
    float t = 0.0f;
#pragma unroll
    for (int w = 0; w < 8; ++w) t += sRed[w];
    out[0] = t * (1.0f / ((float)NB * (float)HH * (float)WW));
  }
}

extern "C" void kernel_launch(void* const* d_in, const int* in_sizes, int n_in,
                              void* d_out, int out_size, void* d_ws, size_t ws_size,
                              hipStream_t stream) {
  (void)in_sizes; (void)n_in; (void)out_size; (void)ws_size;
  const float* real = (const float*)d_in[0];
  const float* fake = (const float*)d_in[1];
  float* partial = (float*)d_ws;  // 4096 floats = 16 KB of scratch
  dim3 grid(WW / TILE, HH / TILE, NB);  // 16 x 16 x 16
  darkchannel_mse_tile<<<grid, NTHREADS, 0, stream>>>(real, fake, partial);
  darkchannel_mse_finish<<<1, NTHREADS, 0, stream>>>(partial, (float*)d_out);
}